// CRD_65446711657116
// MI455X (gfx1250) — compile-verified
//
#include <hip/hip_runtime.h>
#include <hip/hip_bf16.h>

typedef __attribute__((ext_vector_type(2))) float v2f;
typedef __attribute__((ext_vector_type(8))) float v8f;

#define N_NODES 50000
#define N_EDGES 800000
#define D_IN    512
#define D_OUT   256

// ---------------------------------------------------------------------------
// Kernel 1: xw = x @ W  via V_WMMA_F32_16X16X4_F32 (full fp32 matrix path).
// One 16x16 output tile per wave. 8 waves per 256-thread block.
// Tiles: 3125 (M) x 16 (N) = 50000 waves = 6250 blocks exactly -> EXEC all-1s.
// ---------------------------------------------------------------------------
__global__ void gcn_gemm_wmma(const float* __restrict__ X,
                              const float* __restrict__ W,
                              float* __restrict__ XW) {
    const int wave = threadIdx.x >> 5;
    const int lane = threadIdx.x & 31;
    const int tile = blockIdx.x * (blockDim.x >> 5) + wave;

    const int tilesN = D_OUT / 16;            // 16
    const int tm = tile / tilesN;             // 0..3124
    const int tn = tile - tm * tilesN;        // 0..15
    if (tm >= N_NODES / 16) return;           // never taken (exact fit), wave-uniform

    const int m0 = tm * 16;
    const int n0 = tn * 16;

    const int half = lane >> 4;               // 0: lanes 0-15, 1: lanes 16-31
    const int l    = lane & 15;

    // A-matrix 16x4 f32 layout: lanes 0-15 row=l hold K={k0,k0+1},
    //                           lanes 16-31 row=l hold K={k0+2,k0+3}.
    const float* Arow = X + (size_t)(m0 + l) * D_IN;

    v8f acc = {};
    #pragma unroll 4
    for (int k0 = 0; k0 < D_IN; k0 += 4) {
        const int ka = k0 + half * 2;                 // even -> 8B aligned
        v2f a = *(const v2f*)(Arow + ka);

        // B-matrix 4x16 f32: VGPR0 = row (k0+2*half), VGPR1 = row (k0+2*half+1),
        // column n0+l striped across lanes (row-striped-across-VGPRs, mirroring A).
        v2f bf;
        bf.x = W[(size_t)(ka + 0) * D_OUT + n0 + l];
        bf.y = W[(size_t)(ka + 1) * D_OUT + n0 + l];

        acc = __builtin_amdgcn_wmma_f32_16x16x4_f32(
            /*neg_a=*/false, a, /*neg_b=*/false, bf,
            /*c_mod=*/(short)0, acc, /*reuse_a=*/false, /*reuse_b=*/false);
    }

    // C/D 16x16 f32 layout: VGPR i, lanes 0-15 -> M=i, lanes 16-31 -> M=8+i; N=l.
    float* outp = XW + (size_t)(m0 + half * 8) * D_OUT + n0 + l;
    #pragma unroll
    for (int i = 0; i < 8; ++i)
        outp[(size_t)i * D_OUT] = acc[i];
}

// ---------------------------------------------------------------------------
// Degree pipeline (self-loop included): deg = 1 + in-degree, dinv = rsqrt(deg)
// ---------------------------------------------------------------------------
__global__ void gcn_deg_init(float* __restrict__ deg) {
    int i = blockIdx.x * blockDim.x + threadIdx.x;
    if (i < N_NODES) deg[i] = 1.0f;               // self-loop
}

__global__ void gcn_deg_accum(const long long* __restrict__ ei,
                              float* __restrict__ deg) {
    int e = blockIdx.x * blockDim.x + threadIdx.x;
    if (e < N_EDGES) {
        long long col = ei[(size_t)N_EDGES + e];  // targets
        atomicAdd(&deg[col], 1.0f);
    }
}

__global__ void gcn_dinv(const float* __restrict__ deg,
                         float* __restrict__ dinv) {
    int i = blockIdx.x * blockDim.x + threadIdx.x;
    if (i < N_NODES) dinv[i] = rsqrtf(deg[i]);    // deg >= 1 always
}

// ---------------------------------------------------------------------------
// out[i] = dinv[i]^2 * xw[i]   (self-loop term; also zero-initializes out)
// float4 vectorized: 64 float4 per row.
// ---------------------------------------------------------------------------
__global__ void gcn_self_init(const float* __restrict__ XW,
                              const float* __restrict__ dinv,
                              float* __restrict__ out) {
    int i = blockIdx.x * blockDim.x + threadIdx.x;     // float4 index
    if (i >= N_NODES * (D_OUT / 4)) return;
    int row = i >> 6;                                  // / (256/4)
    float s = dinv[row];
    s = s * s;
    float4 v = ((const float4*)XW)[i];
    v.x *= s; v.y *= s; v.z *= s; v.w *= s;
    ((float4*)out)[i] = v;
}

// ---------------------------------------------------------------------------
// Edge scatter: out[col] += xw[row] * dinv[row]*dinv[col]
// 64 threads per edge; each thread owns one float4 of the 256-wide row.
// xw (51 MB) is L2-resident; atomics resolve in L2.
// ---------------------------------------------------------------------------
__global__ void gcn_scatter(const long long* __restrict__ ei,
                            const float* __restrict__ XW,
                            const float* __restrict__ dinv,
                            float* __restrict__ out) {
    const int epb = 256 / 64;                                  // 4 edges/block
    long long e = (long long)blockIdx.x * epb + (threadIdx.x >> 6);
    if (e >= N_EDGES) return;
    const int t = threadIdx.x & 63;                            // 0..63

    long long row = ei[e];                                     // source
    long long col = ei[(size_t)N_EDGES + e];                   // target
    float norm = dinv[row] * dinv[col];

    float4 v = *(const float4*)(XW + (size_t)row * D_OUT + t * 4);
    float* dst = out + (size_t)col * D_OUT + t * 4;
    atomicAdd(dst + 0, v.x * norm);
    atomicAdd(dst + 1, v.y * norm);
    atomicAdd(dst + 2, v.z * norm);
    atomicAdd(dst + 3, v.w * norm);
}

// ---------------------------------------------------------------------------
// Finalize: out = relu(out + b)
// ---------------------------------------------------------------------------
__global__ void gcn_finalize(float* __restrict__ out,
                             const float* __restrict__ b) {
    int i = blockIdx.x * blockDim.x + threadIdx.x;     // float4 index
    if (i >= N_NODES * (D_OUT / 4)) return;
    int c4 = (i & 63);                                 // float4 column
    float4 v  = ((float4*)out)[i];
    float4 bb = ((const float4*)b)[c4];
    v.x = fmaxf(v.x + bb.x, 0.0f);
    v.y = fmaxf(v.y + bb.y, 0.0f);
    v.z = fmaxf(v.z + bb.z, 0.0f);
    v.w = fmaxf(v.w + bb.w, 0.0f);
    ((float4*)out)[i] = v;
}

// ---------------------------------------------------------------------------
extern "C" void kernel_launch(void* const* d_in, const int* in_sizes, int n_in,
                              void* d_out, int out_size, void* d_ws, size_t ws_size,
                              hipStream_t stream) {
    const float*     x  = (const float*)d_in[0];       // [50000, 512]
    const long long* ei = (const long long*)d_in[1];   // [2, 800000] int64
    const float*     W  = (const float*)d_in[2];       // [512, 256]
    const float*     b  = (const float*)d_in[3];       // [256]
    float*           out = (float*)d_out;              // [50000, 256]

    // Workspace layout (bytes):
    //   xw  : 50000*256*4 = 51,200,000
    //   deg : 50000*4
    //   dinv: 50000*4
    char* ws = (char*)d_ws;
    float* xw   = (float*)ws;
    float* deg  = (float*)(ws + ((size_t)N_NODES * D_OUT * 4 + 255 & ~(size_t)255));
    float* dinv = deg + N_NODES;

    // 1) GEMM: 50000 waves (= tiles), 8 waves per block -> 6250 blocks
    gcn_gemm_wmma<<<(N_NODES / 16) * (D_OUT / 16) / 8, 256, 0, stream>>>(x, W, xw);

    // 2) degrees
    gcn_deg_init <<<(N_NODES + 255) / 256, 256, 0, stream>>>(deg);
    gcn_deg_accum<<<(N_EDGES + 255) / 256, 256, 0, stream>>>(ei, deg);
    gcn_dinv     <<<(N_NODES + 255) / 256, 256, 0, stream>>>(deg, dinv);

    // 3) self-loop init of out (also clears poison)
    const int nvec4 = N_NODES * (D_OUT / 4);           // 3,200,000
    gcn_self_init<<<(nvec4 + 255) / 256, 256, 0, stream>>>(xw, dinv, out);

    // 4) edge scatter (4 edges per 256-thread block)
    gcn_scatter<<<(N_EDGES + 3) / 4, 256, 0, stream>>>(ei, xw, dinv, out);

    // 5) bias + relu
    gcn_finalize<<<(nvec4 + 255) / 256, 256, 0, stream>>>(out, b);
}